// SimilarityModule_36447092474557
// MI455X (gfx1250) — compile-verified
//
#include <hip/hip_runtime.h>
#include <math.h>

// MI455X / gfx1250, wave32. FP32 WMMA (16x16x4) fused MaxSim GEMM with
// double-buffered GLOBAL_LOAD_ASYNC_TO_LDS staging (ASYNCcnt pipeline).
//
// Workspace layout (floats):
//   [0      .. 8192)   inv-norms of concept rows (T*Q = 8192)
//   [8192   .. 24576)  inv-norms of frame rows   (V*F = 16384)
//   [24576  .. 90112)  l_sim [256][256]
// Total 360448 bytes.

typedef __attribute__((ext_vector_type(2))) float v2f;
typedef __attribute__((ext_vector_type(4))) float v4f;
typedef __attribute__((ext_vector_type(8))) float v8f;

#define SM_T 256
#define SM_V 256
#define SM_Q 32
#define SM_F 64
#define SM_D 512
#define SM_ALPHA 0.7f

// ---------------------------------------------------------------------------
// Kernel 1: per-row inverse L2 norms for concept (8192 rows) and frame
// (16384 rows). One wave32 per row; coalesced strided loads + xor-shuffle
// reduction.
// ---------------------------------------------------------------------------
__global__ void __launch_bounds__(256)
norms_kernel(const float* __restrict__ concept_e,
             const float* __restrict__ frame_e,
             float* __restrict__ invc,
             float* __restrict__ invf) {
  const int lane = threadIdx.x & 31;
  const int wv = blockIdx.x * (blockDim.x >> 5) + (threadIdx.x >> 5);
  const float* src;
  float* dst;
  if (wv < SM_T * SM_Q) {
    src = concept_e + (size_t)wv * SM_D;
    dst = invc + wv;
  } else {
    const int r = wv - SM_T * SM_Q;  // < 16384 by grid sizing
    src = frame_e + (size_t)r * SM_D;
    dst = invf + r;
  }
  float s = 0.0f;
  for (int i = lane; i < SM_D; i += 32) {
    const float x = src[i];
    s = fmaf(x, x, s);
  }
#pragma unroll
  for (int off = 16; off > 0; off >>= 1) s += __shfl_xor(s, off, 32);
  if (lane == 0) *dst = 1.0f / fmaxf(sqrtf(s), 1e-12f);
}

// ---------------------------------------------------------------------------
// Kernel 2: fused scores GEMM + (max over F) + (mean over Q).
// Grid: (T/2, V). Block: 256 threads = 8 waves.
// Each WG computes the raw 64x64 tile scoresRaw[m][f], K = 512 staged in LDS
// chunks of 64 via double-buffered global_load_async_to_lds_b128:
//   - chunk c+1 is issued before waiting on chunk c (async loads complete
//     in order per wave, so s_wait_asynccnt 8 <=> chunk c resident),
//   - end-of-iteration barrier gates the WAR on buffer reuse.
// Rows padded to 68 floats: fragment float2 loads hit banks {4r+k..4r+k+3}
// -> all 64 banks exactly once per wave.
// ---------------------------------------------------------------------------
__global__ void __launch_bounds__(256)
maxsim_kernel(const float* __restrict__ concept_e,
              const float* __restrict__ frame_e,
              const float* __restrict__ invc,
              const float* __restrict__ invf,
              float* __restrict__ lsim) {
  __shared__ float As[2][64][68];
  __shared__ float Bs[2][64][68];
  __shared__ float rowv[64];

  const int tid = threadIdx.x;
  const int lane = tid & 31;
  const int wave = tid >> 5;
  const int mt = wave & 3;           // 0..3 : 16-row band of M
  const int nt0 = (wave >> 2) << 1;  // 0 or 2 : first 16-col band of N
  const int tpair = blockIdx.x;      // tokens 2*tpair, 2*tpair+1
  const int vblk = blockIdx.y;       // video index

  const float* Ag = concept_e + (size_t)tpair * 64 * SM_D;
  const float* Bg = frame_e + (size_t)vblk * 64 * SM_D;

  // Cooperative mapping: 1024 float4 segments per array, 4 per thread.
  // Issue 8 async memory->LDS copies per wave per chunk (no VGPR round-trip).
  auto issue_chunk = [&](int chunk) {
    const int k0 = chunk * 64;
    const int buf = chunk & 1;
#pragma unroll
    for (int i = 0; i < 4; ++i) {
      const int idx = tid + i * 256;  // 0..1023
      const int row = idx >> 4;
      const int kk = (idx & 15) << 2;
      // Flat shared-memory addresses carry the LDS byte offset in [31:0].
      const unsigned la = (unsigned)(size_t)&As[buf][row][kk];
      const unsigned lb = (unsigned)(size_t)&Bs[buf][row][kk];
      const float* ga = Ag + row * SM_D + k0 + kk;
      const float* gb = Bg + row * SM_D + k0 + kk;
      asm volatile("global_load_async_to_lds_b128 %0, %1, off"
                   :: "v"(la), "v"(ga) : "memory");
      asm volatile("global_load_async_to_lds_b128 %0, %1, off"
                   :: "v"(lb), "v"(gb) : "memory");
    }
  };

  v8f acc0 = {};
  v8f acc1 = {};

  const int lrow = lane & 15;
  const int koff = (lane >> 4) << 1;  // 0 or 2 (f32 WMMA A/B lane layout)

  issue_chunk(0);

  for (int c = 0; c < 8; ++c) {
    const int buf = c & 1;
    if (c + 1 < 8) {
      issue_chunk(c + 1);  // prefetch next chunk into the other buffer
      asm volatile("s_wait_asynccnt 0x8" ::: "memory");  // chunk c landed
    } else {
      asm volatile("s_wait_asynccnt 0x0" ::: "memory");
    }
    __syncthreads();  // chunk c resident for every wave

    const float* ap = &As[buf][mt * 16 + lrow][koff];
    const float* bp0 = &Bs[buf][nt0 * 16 + lrow][koff];
    const float* bp1 = bp0 + 16 * 68;
#pragma unroll
    for (int k = 0; k < 64; k += 4) {
      const v2f a = *(const v2f*)(ap + k);
      const v2f b0 = *(const v2f*)(bp0 + k);
      const v2f b1 = *(const v2f*)(bp1 + k);
      acc0 = __builtin_amdgcn_wmma_f32_16x16x4_f32(false, a, false, b0,
                                                   (short)0, acc0, false, false);
      acc1 = __builtin_amdgcn_wmma_f32_16x16x4_f32(false, a, false, b1,
                                                   (short)0, acc1, false, false);
    }

    __syncthreads();  // all waves done reading buf before it is re-filled
  }

  // Epilogue: write scores * invn_f into LDS (reuse As[0] as [64][68]).
  // C/D layout: VGPR j holds M = mt*16 + j + 8*(lane>>4), N = nt*16 + (lane&15).
  float* Ss = &As[0][0][0];
  const int nl0 = nt0 * 16 + lrow;
  const int nl1 = nl0 + 16;
  const float if0 = invf[vblk * 64 + nl0];
  const float if1 = invf[vblk * 64 + nl1];
  const int mbase = mt * 16 + ((lane >> 4) << 3);
#pragma unroll
  for (int j = 0; j < 8; ++j) {
    Ss[(mbase + j) * 68 + nl0] = acc0[j] * if0;
    Ss[(mbase + j) * 68 + nl1] = acc1[j] * if1;
  }
  __syncthreads();

  // Row max over the 64 frames, scaled by invn_c of that (token, q) row.
  if (tid < 64) {
    const float* r = Ss + tid * 68;
    float mx = r[0];
#pragma unroll
    for (int f = 1; f < 64; ++f) mx = fmaxf(mx, r[f]);
    rowv[tid] = mx * invc[tpair * 64 + tid];
  }
  __syncthreads();

  // Mean over the 32 q's of each token.
  if (tid < 2) {
    float s = 0.0f;
#pragma unroll
    for (int q = 0; q < 32; ++q) s += rowv[tid * 32 + q];
    lsim[(size_t)(tpair * 2 + tid) * SM_V + vblk] = s * (1.0f / 32.0f);
  }
}

// ---------------------------------------------------------------------------
// Kernel 3: out[t][v] = scale * (ALPHA * dot(text[t], clip[v]) +
//                                (1-ALPHA) * lsim[t][v])
// Tiny GEMM (67 MFLOP), fully L2-resident; plain FMA loop with float4 loads.
// ---------------------------------------------------------------------------
__global__ void __launch_bounds__(256)
combine_kernel(const float* __restrict__ text_e,
               const float* __restrict__ clip_e,
               const float* __restrict__ lsim,
               const float* __restrict__ logit_scale,
               float* __restrict__ out) {
  const int t = blockIdx.x;
  const int v = threadIdx.x;
  const float scale = fminf(expf(logit_scale[0]), 100.0f);
  const float* tr = text_e + (size_t)t * SM_D;
  const float* cr = clip_e + (size_t)v * SM_D;
  float s = 0.0f;
#pragma unroll 4
  for (int d = 0; d < SM_D; d += 4) {
    const v4f a = *(const v4f*)(tr + d);
    const v4f b = *(const v4f*)(cr + d);
    s = fmaf(a.x, b.x, s);
    s = fmaf(a.y, b.y, s);
    s = fmaf(a.z, b.z, s);
    s = fmaf(a.w, b.w, s);
  }
  out[(size_t)t * SM_V + v] =
      scale * (SM_ALPHA * s + (1.0f - SM_ALPHA) * lsim[(size_t)t * SM_V + v]);
}

// ---------------------------------------------------------------------------
extern "C" void kernel_launch(void* const* d_in, const int* in_sizes, int n_in,
                              void* d_out, int out_size, void* d_ws,
                              size_t ws_size, hipStream_t stream) {
  const float* text_e = (const float*)d_in[0];     // [256, 512]
  const float* clip_e = (const float*)d_in[1];     // [256, 512]
  const float* concept_e = (const float*)d_in[2];  // [256, 32, 512]
  const float* frame_e = (const float*)d_in[3];    // [256, 64, 512]
  const float* logit_scale = (const float*)d_in[4];

  float* ws = (float*)d_ws;
  float* invc = ws;                  // 8192
  float* invf = ws + 8192;           // 16384
  float* lsim = ws + 24576;          // 65536
  float* out = (float*)d_out;        // [256, 256]

  // 24576 rows total, 8 waves (rows) per 256-thread block -> 3072 blocks.
  norms_kernel<<<3072, 256, 0, stream>>>(concept_e, frame_e, invc, invf);

  dim3 grid(SM_T / 2, SM_V);  // (128, 256)
  maxsim_kernel<<<grid, 256, 0, stream>>>(concept_e, frame_e, invc, invf, lsim);

  combine_kernel<<<SM_T, SM_V, 0, stream>>>(text_e, clip_e, lsim, logit_scale,
                                            out);
}